// MultiBoxLoss_28630251995354
// MI455X (gfx1250) — compile-verified
//
#include <hip/hip_runtime.h>
#include <math.h>

#define TPB 256

typedef __attribute__((ext_vector_type(2))) float v2f;
typedef __attribute__((ext_vector_type(8))) float v8f;

__device__ __forceinline__ float smooth_l1(float x){
  float a = fabsf(x);
  return (a < 1.f) ? 0.5f * a * a : a - 0.5f;
}

// pack (iou, prior) so that u64-max == (max iou, then lowest prior index) == jnp.argmax tie-break
__device__ __forceinline__ unsigned long long pack_ov(float v, unsigned p){
  return ((unsigned long long)__float_as_uint(v) << 32) |
         (unsigned long long)(0xFFFFFFFFu - p);
}

// ---------------------------------------------------------------- K0: zero bp
__global__ void mbox_k0_zero(unsigned long long* __restrict__ p, int n){
  int i = blockIdx.x * blockDim.x + threadIdx.x;
  if (i < n) p[i] = 0ull;
}

// ---------------------------------------------------------------- K1: matching
// grid (nblk, B), 256 thr. Per-prior best over T; per-truth best over P via
// wave shfl pre-max + LDS u64 atomicMax + global u64 atomicMax (deterministic).
__global__ void mbox_k1_match(const float* __restrict__ pri, const float* __restrict__ tgt,
                              unsigned long long* __restrict__ bp,
                              float* __restrict__ bto, int* __restrict__ bti,
                              int P, int T){
  extern __shared__ unsigned char smem[];
  unsigned long long* ldsBP = (unsigned long long*)smem;
  float* sT = (float*)(smem + (size_t)T * 8);
  int b = blockIdx.y;
  int tid = threadIdx.x;
  for (int i = tid; i < T; i += TPB) ldsBP[i] = 0ull;
  for (int i = tid; i < T * 15; i += TPB) sT[i] = tgt[(size_t)b * T * 15 + i];
  __syncthreads();

  int p = blockIdx.x * TPB + tid;
  float px1 = 0.f, py1 = 0.f, px2 = 0.f, py2 = 0.f, areaP = 0.f;
  if (p < P){
    float4 pr = ((const float4*)pri)[p];
    px1 = pr.x - 0.5f * pr.z; py1 = pr.y - 0.5f * pr.w;
    px2 = pr.x + 0.5f * pr.z; py2 = pr.y + 0.5f * pr.w;
    areaP = (px2 - px1) * (py2 - py1);
  }
  float best = -1.f; int besti = 0;
  for (int t = 0; t < T; ++t){
    unsigned long long pk = 0ull;
    if (p < P){
      float tx1 = sT[t*15+0], ty1 = sT[t*15+1], tx2 = sT[t*15+2], ty2 = sT[t*15+3];
      float iw = fmaxf(fminf(tx2, px2) - fmaxf(tx1, px1), 0.f);
      float ih = fmaxf(fminf(ty2, py2) - fmaxf(ty1, py1), 0.f);
      float inter = iw * ih;
      float areaT = (tx2 - tx1) * (ty2 - ty1);
      float iou = inter / (areaT + areaP - inter);
      if (iou > best){ best = iou; besti = t; }   // strict > keeps first t (argmax semantics)
      pk = pack_ov(iou, (unsigned)p);
    }
    // wave32 pre-max (manual 2x32-bit shuffles), then 1 LDS atomic per wave
    for (int off = 16; off > 0; off >>= 1){
      unsigned hi = (unsigned)(pk >> 32), lo = (unsigned)pk;
      hi = __shfl_down(hi, off); lo = __shfl_down(lo, off);
      unsigned long long o = ((unsigned long long)hi << 32) | lo;
      if (o > pk) pk = o;
    }
    if ((tid & 31) == 0) atomicMax(&ldsBP[t], pk);
  }
  if (p < P){
    size_t idx = (size_t)b * P + p;
    bto[idx] = best;
    bti[idx] = besti;
  }
  __syncthreads();
  for (int t = tid; t < T; t += TPB)
    atomicMax(&bp[(size_t)b * T + t], ldsBP[t]);
}

// ---------------------------------------------------------------- K2: scatter fixups
// Serial ascending loop reproduces scatter-max + last-write-wins set exactly.
__global__ void mbox_k2_fix(const unsigned long long* __restrict__ bp,
                            float* __restrict__ bto, int* __restrict__ bti,
                            int* __restrict__ anyv, int P, int T){
  if (threadIdx.x != 0) return;
  int b = blockIdx.x;
  int any = 0;
  for (int t = 0; t < T; ++t){
    unsigned long long pk = bp[(size_t)b * T + t];
    float ov = __uint_as_float((unsigned)(pk >> 32));
    unsigned j = 0xFFFFFFFFu - (unsigned)(pk & 0xFFFFFFFFull);
    size_t idx = (size_t)b * P + j;
    if (ov >= 0.2f){ bto[idx] = fmaxf(bto[idx], 2.0f); any = 1; }
    bti[idx] = (int)t;                       // unconditional set, last t wins
  }
  anyv[b] = any;
}

// ---------------------------------------------------------------- K3: per-prior losses
__global__ void mbox_k3_losses(const float* __restrict__ loc, const float* __restrict__ cf,
                               const float* __restrict__ lnd, const float* __restrict__ pri,
                               const float* __restrict__ tgt,
                               const float* __restrict__ bto, const int* __restrict__ bti,
                               const int* __restrict__ anyv,
                               float* __restrict__ rank,
                               float* __restrict__ ll_part, float* __restrict__ lm_part,
                               float* __restrict__ pos_part, float* __restrict__ pos1_part,
                               float* __restrict__ cep_part,
                               int P, int T, int nblk){
  __shared__ float sh[TPB];
  int b = blockIdx.y;
  int tid = threadIdx.x;
  int p = blockIdx.x * TPB + tid;
  float llv = 0.f, lmv = 0.f, posf = 0.f, pos1f = 0.f, cep = 0.f;
  if (p < P){
    size_t idx = (size_t)b * P + p;
    __builtin_prefetch((const void*)(lnd + idx * 10), 0, 1);  // -> global_prefetch_b8
    __builtin_prefetch((const void*)(cf + idx * 2), 0, 1);
    int t = bti[idx];
    float btov = bto[idx];
    int any = anyv[b];
    const float* tr = tgt + ((size_t)b * T + t) * 15;
    float4 pr = ((const float4*)pri)[p];
    float conf_f = (btov < 0.35f) ? 0.f : tr[14];
    if (!any) conf_f = 0.f;
    int ci = (int)conf_f;
    bool pos = (ci != 0);
    bool pos1 = (ci > 0);
    if (pos){
      float mx1 = tr[0], my1 = tr[1], mx2 = tr[2], my2 = tr[3];
      float lt0 = ((mx1 + mx2) * 0.5f - pr.x) / (0.1f * pr.z);
      float lt1 = ((my1 + my2) * 0.5f - pr.y) / (0.1f * pr.w);
      float lt2 = logf((mx2 - mx1) / pr.z) / 0.2f;
      float lt3 = logf((my2 - my1) / pr.w) / 0.2f;
      const float* ld = loc + idx * 4;
      llv = smooth_l1(ld[0] - lt0) + smooth_l1(ld[1] - lt1)
          + smooth_l1(ld[2] - lt2) + smooth_l1(ld[3] - lt3);
    }
    if (pos1){
      const float* lv = lnd + idx * 10;
      float s = 0.f;
      #pragma unroll
      for (int k = 0; k < 5; ++k){
        float ltx = (tr[4 + 2*k]     - pr.x) / (0.1f * pr.z);
        float lty = (tr[4 + 2*k + 1] - pr.y) / (0.1f * pr.w);
        s += smooth_l1(lv[2*k] - ltx) + smooth_l1(lv[2*k + 1] - lty);
      }
      lmv = s;
    }
    float x0 = cf[idx*2 + 0], x1 = cf[idx*2 + 1];
    float mx = fmaxf(x0, x1);
    float lse = mx + logf(expf(x0 - mx) + expf(x1 - mx));
    float cev = lse - (pos ? x1 : x0);          // target = pos?1:0
    rank[idx] = pos ? 0.f : cev;                // rank_loss (>=0)
    if (pos) cep = cev;
    posf  = pos  ? 1.f : 0.f;
    pos1f = pos1 ? 1.f : 0.f;
  }
  int blk = b * nblk + blockIdx.x;
  float vals[5] = {llv, lmv, posf, pos1f, cep};
  float* outs[5] = {ll_part, lm_part, pos_part, pos1_part, cep_part};
  for (int q = 0; q < 5; ++q){
    sh[tid] = vals[q]; __syncthreads();
    for (int s = TPB / 2; s > 0; s >>= 1){
      if (tid < s) sh[tid] += sh[tid + s];
      __syncthreads();
    }
    if (tid == 0) outs[q][blk] = sh[0];
    __syncthreads();
  }
}

// ---------------------------------------------------------------- K4: single-wave WMMA reduction
// Sums n floats with chained V_WMMA_F32_16X16X4_F32: A=16x4 chunk, B=ones,
// C accumulates per-row sums. Exact f32 MACs, fixed order -> deterministic.
// Full 64-element chunks use one unconditional global_load_b64 per lane
// (A[m][kh], A[m][kh+1] are contiguous) -> no divergence around the WMMA.
__device__ float wave_wmma_sum32(const float* __restrict__ src, int n){
  int lane = threadIdx.x & 31;
  int m = lane & 15;
  int kh = (lane >> 4) << 1;        // K-half: 0 or 2
  v2f ones = {1.f, 1.f};
  v8f acc = {};
  int nfull = n & ~63;
  const float* lp = src + m * 4 + kh;
  for (int base = 0; base < nfull; base += 64){
    float2 d = *(const float2*)(lp + base);   // global_load_b64, unconditional
    v2f a; a[0] = d.x; a[1] = d.y;
    acc = __builtin_amdgcn_wmma_f32_16x16x4_f32(
        false, a, false, ones, (short)0, acc, false, false);
  }
  if (nfull < n){                    // uniform tail branch (dead when n % 64 == 0)
    v2f a; a[0] = 0.f; a[1] = 0.f;
    int v0 = nfull + m * 4 + kh;
    if (v0 < n)     a[0] = src[v0];
    if (v0 + 1 < n) a[1] = src[v0 + 1];
    acc = __builtin_amdgcn_wmma_f32_16x16x4_f32(
        false, a, false, ones, (short)0, acc, false, false);
  }
  // lane 0 holds D[M=0..7][0], lane 16 holds D[M=8..15][0]
  float s = acc[0]+acc[1]+acc[2]+acc[3]+acc[4]+acc[5]+acc[6]+acc[7];
  return __shfl(s, 0) + __shfl(s, 16);
}

__global__ void mbox_k4_reduce(const float* __restrict__ ll_part, const float* __restrict__ lm_part,
                               const float* __restrict__ pos_part, const float* __restrict__ pos1_part,
                               const float* __restrict__ cep_part,
                               int* __restrict__ numpos, int* __restrict__ numneg,
                               float* __restrict__ cepos_sum, float* __restrict__ scal,
                               int B, int nblk, int P){
  int lane = threadIdx.x;
  for (int b = lane; b < B; b += 32){
    float np = 0.f, cp = 0.f;
    const float* pp = pos_part + (size_t)b * nblk;
    const float* cc = cep_part + (size_t)b * nblk;
    for (int i = 0; i < nblk; ++i){ np += pp[i]; cp += cc[i]; }
    int npi = (int)(np + 0.5f);
    numpos[b] = npi;
    long long nn = 7LL * npi, cap = (long long)P - 1;
    numneg[b] = (int)(nn < cap ? nn : cap);
    cepos_sum[b] = cp;
  }
  int NB1 = B * nblk;               // all 32 lanes reconverged here (EXEC all 1s for WMMA)
  float llv = wave_wmma_sum32(ll_part, NB1);
  float lmv = wave_wmma_sum32(lm_part, NB1);
  float p1v = wave_wmma_sum32(pos1_part, NB1);
  float npv = wave_wmma_sum32(pos_part, NB1);
  if (lane == 0){ scal[0] = llv; scal[1] = lmv; scal[2] = p1v; scal[3] = npv; }
}

// ---------------------------------------------------------------- K5: top-K sum via 4-pass radix select
// keys = float bits (all values >= 0 -> monotonic as u32). Exact K-th largest
// threshold t, count R of t-copies to take; sum = sum(v > t) + R*t.
__global__ void mbox_k5_topk(const float* __restrict__ rank, const int* __restrict__ numneg,
                             float* __restrict__ topk, int P){
  __shared__ unsigned hist[256];
  __shared__ unsigned s_pref;
  __shared__ int s_R;
  __shared__ float red[TPB];
  int b = blockIdx.x;
  int tid = threadIdx.x;
  const float* r = rank + (size_t)b * P;
  int K = numneg[b];
  if (K <= 0){ if (tid == 0) topk[b] = 0.f; return; }   // uniform branch
  unsigned pref = 0; int R = K;
  for (int byte = 3; byte >= 0; --byte){
    hist[tid] = 0;
    __syncthreads();
    unsigned himask = (byte == 3) ? 0u : (0xFFFFFFFFu << ((byte + 1) * 8));
    for (int p = tid; p < P; p += TPB){
      unsigned key = __float_as_uint(r[p]);
      if ((key & himask) == pref)
        atomicAdd(&hist[(key >> (byte * 8)) & 0xFF], 1u);
    }
    __syncthreads();
    if (tid == 0){
      int rr = R; unsigned bin = 0;
      for (int i = 255; i >= 0; --i){
        int c = (int)hist[i];
        if (rr > c) rr -= c;
        else { bin = (unsigned)i; break; }
      }
      s_pref = pref | (bin << (byte * 8));
      s_R = rr;
    }
    __syncthreads();
    pref = s_pref; R = s_R;
    __syncthreads();
  }
  float tval = __uint_as_float(pref);
  float part = 0.f;
  for (int p = tid; p < P; p += TPB){
    unsigned key = __float_as_uint(r[p]);
    if (key > pref) part += r[p];
  }
  red[tid] = part; __syncthreads();
  for (int s = TPB / 2; s > 0; s >>= 1){
    if (tid < s) red[tid] += red[tid + s];
    __syncthreads();
  }
  if (tid == 0) topk[b] = red[0] + (float)R * tval;
}

// ---------------------------------------------------------------- K6: finalize
__global__ void mbox_k6_final(const float* __restrict__ cepos_sum, const float* __restrict__ topk,
                              const float* __restrict__ scal, float* __restrict__ out, int B){
  if (threadIdx.x != 0 || blockIdx.x != 0) return;
  float lc = 0.f;
  for (int b = 0; b < B; ++b) lc += cepos_sum[b] + topk[b];
  float N  = fmaxf(scal[3], 1.f);
  float N1 = fmaxf(scal[2], 1.f);
  out[0] = scal[0] / N;   // loss_l / N
  out[1] = lc / N;        // loss_c / N
  out[2] = scal[1] / N1;  // loss_landm / N1
}

// ---------------------------------------------------------------- host
extern "C" void kernel_launch(void* const* d_in, const int* in_sizes, int n_in,
                              void* d_out, int out_size, void* d_ws, size_t ws_size,
                              hipStream_t stream){
  const float* loc = (const float*)d_in[0];
  const float* cf  = (const float*)d_in[1];
  const float* lnd = (const float*)d_in[2];
  const float* pri = (const float*)d_in[3];
  const float* tgt = (const float*)d_in[4];
  int P = in_sizes[3] / 4;
  int B = in_sizes[0] / (4 * P);
  int T = in_sizes[4] / (15 * B);
  int nblk = (P + TPB - 1) / TPB;
  int NB1 = B * nblk;
  (void)n_in; (void)out_size; (void)ws_size;

  char* ws = (char*)d_ws;
  size_t off = 0;
  auto carve = [&](size_t bytes) -> char* {
    char* r = ws + off;
    off += (bytes + 15) & ~((size_t)15);
    return r;
  };
  unsigned long long* bp = (unsigned long long*)carve((size_t)B * T * 8);
  float* bto       = (float*)carve((size_t)B * P * 4);
  int*   bti       = (int*)  carve((size_t)B * P * 4);
  float* rank      = (float*)carve((size_t)B * P * 4);
  int*   anyv      = (int*)  carve((size_t)B * 4);
  float* ll_part   = (float*)carve((size_t)NB1 * 4);
  float* lm_part   = (float*)carve((size_t)NB1 * 4);
  float* pos_part  = (float*)carve((size_t)NB1 * 4);
  float* pos1_part = (float*)carve((size_t)NB1 * 4);
  float* cep_part  = (float*)carve((size_t)NB1 * 4);
  int*   numpos    = (int*)  carve((size_t)B * 4);
  int*   numneg    = (int*)  carve((size_t)B * 4);
  float* cepos_sum = (float*)carve((size_t)B * 4);
  float* topk      = (float*)carve((size_t)B * 4);
  float* scal      = (float*)carve(4 * 4);

  mbox_k0_zero<<<(B * T + TPB - 1) / TPB, TPB, 0, stream>>>(bp, B * T);
  size_t sm1 = (size_t)T * 8 + (size_t)T * 15 * 4;
  mbox_k1_match<<<dim3(nblk, B), TPB, sm1, stream>>>(pri, tgt, bp, bto, bti, P, T);
  mbox_k2_fix<<<B, 1, 0, stream>>>(bp, bto, bti, anyv, P, T);
  mbox_k3_losses<<<dim3(nblk, B), TPB, 0, stream>>>(loc, cf, lnd, pri, tgt, bto, bti, anyv,
                                                    rank, ll_part, lm_part, pos_part,
                                                    pos1_part, cep_part, P, T, nblk);
  mbox_k4_reduce<<<1, 32, 0, stream>>>(ll_part, lm_part, pos_part, pos1_part, cep_part,
                                       numpos, numneg, cepos_sum, scal, B, nblk, P);
  mbox_k5_topk<<<B, TPB, 0, stream>>>(rank, numneg, topk, P);
  mbox_k6_final<<<1, 1, 0, stream>>>(cepos_sum, topk, scal, (float*)d_out, B);
}